// Tox21GNN_4475355922840
// MI455X (gfx1250) — compile-verified
//
#include <hip/hip_runtime.h>
#include <hip/hip_bf16.h>

#define N_NODES 131072
#define E_EDGES 262144
#define G_GRAPHS 4096

typedef __attribute__((ext_vector_type(2))) float v2f;
typedef __attribute__((ext_vector_type(8))) float v8f;

// ---------------- degree / dinv ----------------
__global__ void deg_kernel(const int* __restrict__ dst, float* __restrict__ deg, int E) {
    int e = blockIdx.x * blockDim.x + threadIdx.x;
    if (e < E) atomicAdd(&deg[dst[e]], 1.0f);
}

__global__ void dinv_kernel(float* __restrict__ deg, int n) {
    int i = blockIdx.x * blockDim.x + threadIdx.x;
    if (i < n) deg[i] = rsqrtf(deg[i] + 1.0f);
}

// ---------------- layer 1: h = x @ W1 (outer product, fan_in = 1) ----------------
__global__ void layer1_kernel(const float* __restrict__ x, const float* __restrict__ W1,
                              float* __restrict__ h, int n) {
    int tid = blockIdx.x * blockDim.x + threadIdx.x;
    if (tid >= n * 64) return;
    int i = tid >> 6;
    int f = tid & 63;
    h[tid] = x[i] * W1[f];
}

// ---------------- edge scatter: agg[dst] += h[src] * dinv[src]*dinv[dst] ----------------
// One thread per (edge, 16-float chunk): amortizes index/coef loads over 16 features.
__global__ void edge_agg_kernel(const int* __restrict__ src, const int* __restrict__ dst,
                                const float* __restrict__ h, const float* __restrict__ dinv,
                                float* __restrict__ agg, int E, int F) {
    int C = F >> 4;                       // 16-float chunks per edge
    int tid = blockIdx.x * blockDim.x + threadIdx.x;
    if (tid >= E * C) return;
    int e = tid / C;
    int c = tid - e * C;
    int s = src[e], d = dst[e];
    float coef = dinv[s] * dinv[d];
    const float* hp = h + (size_t)s * F + (c << 4);
    float* ap = agg + (size_t)d * F + (c << 4);
#pragma unroll
    for (int j = 0; j < 4; ++j) {
        const float4 v = *(const float4*)(hp + 4 * j);
        atomicAdd(ap + 4 * j + 0, v.x * coef);
        atomicAdd(ap + 4 * j + 1, v.y * coef);
        atomicAdd(ap + 4 * j + 2, v.z * coef);
        atomicAdd(ap + 4 * j + 3, v.w * coef);
    }
}

// ---------------- self-loop + bias (+ optional relu), in place on agg ----------------
__global__ void selfloop_bias_act(const float* __restrict__ h, const float* __restrict__ dinv,
                                  const float* __restrict__ b, float* __restrict__ agg,
                                  int n, int F, int relu) {
    int tid = blockIdx.x * blockDim.x + threadIdx.x;
    if (tid >= n * F) return;
    int i = tid / F;
    int f = tid - i * F;
    float di = dinv[i];
    float v = agg[tid] + h[tid] * di * di + b[f];
    if (relu) v = fmaxf(v, 0.0f);
    agg[tid] = v;
}

// ---------------- f32 WMMA GEMM, 2x2 register blocking -----------------------------------
// out[rows,F] = X[rows,K] @ W[K,F] (+bias)(+relu). One wave per 32x32 output tile:
// 4 accumulators (acc00 acc01 / acc10 acc11), per K-step of 4 we load 2 A fragments +
// 4 B scalars and issue 4 independent V_WMMA_F32_16X16X4_F32 (no RAW chains -> no NOPs).
// Fragment layouts per ISA 7.12.2:
//   A 16x4 f32: lanes 0-15 hold K=0,1 ; lanes 16-31 hold K=2,3 (2 VGPRs).
//   B 4x16 f32: lanes 0-15 hold K=0,1 for col=lane ; lanes 16-31 hold K=2,3.
//   C/D 16x16 f32: element r -> M = r + 8*(lane>=16), N = lane&15.
__global__ void wmma_gemm2x2(const float* __restrict__ X, const float* __restrict__ W,
                             const float* __restrict__ bias, float* __restrict__ out,
                             int rows, int K, int F, int relu) {
    int warp = threadIdx.x >> 5;
    int lane = threadIdx.x & 31;
    int tiles_n = F >> 5;                          // 32-wide tiles
    int tile = blockIdx.x * (blockDim.x >> 5) + warp;
    int total = (rows >> 5) * tiles_n;
    if (tile >= total) return;
    int mt = tile / tiles_n;
    int nt = tile - mt * tiles_n;
    int m0 = mt << 5;
    int n0 = nt << 5;
    int lo = lane & 15;
    int hi = lane >> 4;

    v8f acc00, acc01, acc10, acc11;
    float bv0 = bias ? bias[n0 + lo] : 0.0f;
    float bv1 = bias ? bias[n0 + 16 + lo] : 0.0f;
#pragma unroll
    for (int r = 0; r < 8; ++r) {
        acc00[r] = bv0; acc01[r] = bv1;
        acc10[r] = bv0; acc11[r] = bv1;
    }

    const float* arow0 = X + (size_t)(m0 + lo) * K + 2 * hi;       // rows m0..m0+15
    const float* arow1 = arow0 + (size_t)16 * K;                   // rows m0+16..m0+31
    const float* bc0 = W + (size_t)(2 * hi) * F + n0 + lo;         // cols n0..n0+15
    const float* bc1 = bc0 + 16;                                   // cols n0+16..n0+31

#pragma unroll 2
    for (int kk = 0; kk < K; kk += 4) {
        v2f a0 = *(const v2f*)(arow0 + kk);
        v2f a1 = *(const v2f*)(arow1 + kk);
        v2f b0, b1;
        b0[0] = bc0[(size_t)kk * F];
        b0[1] = bc0[(size_t)(kk + 1) * F];
        b1[0] = bc1[(size_t)kk * F];
        b1[1] = bc1[(size_t)(kk + 1) * F];
        acc00 = __builtin_amdgcn_wmma_f32_16x16x4_f32(false, a0, false, b0, (short)0, acc00, false, false);
        acc01 = __builtin_amdgcn_wmma_f32_16x16x4_f32(false, a0, false, b1, (short)0, acc01, false, false);
        acc10 = __builtin_amdgcn_wmma_f32_16x16x4_f32(false, a1, false, b0, (short)0, acc10, false, false);
        acc11 = __builtin_amdgcn_wmma_f32_16x16x4_f32(false, a1, false, b1, (short)0, acc11, false, false);
    }

#pragma unroll
    for (int r = 0; r < 8; ++r) {
        int mA = m0 + r + 8 * hi;
        int mB = mA + 16;
        float v00 = acc00[r], v01 = acc01[r], v10 = acc10[r], v11 = acc11[r];
        if (relu) {
            v00 = fmaxf(v00, 0.0f); v01 = fmaxf(v01, 0.0f);
            v10 = fmaxf(v10, 0.0f); v11 = fmaxf(v11, 0.0f);
        }
        out[(size_t)mA * F + n0 + lo]      = v00;
        out[(size_t)mA * F + n0 + 16 + lo] = v01;
        out[(size_t)mB * F + n0 + lo]      = v10;
        out[(size_t)mB * F + n0 + 16 + lo] = v11;
    }
}

// ---------------- global mean pool ----------------
__global__ void pool_sum_kernel(const float* __restrict__ h, const int* __restrict__ batch,
                                float* __restrict__ sums, float* __restrict__ cnts, int n) {
    int tid = blockIdx.x * blockDim.x + threadIdx.x;
    if (tid >= n * 64) return;
    int i = tid >> 6;
    int c = tid & 63;
    int g = batch[i];
    const float4 v = *(const float4*)(h + (size_t)i * 256 + (c << 2));
    float* s = sums + (size_t)g * 256 + (c << 2);
    atomicAdd(s + 0, v.x);
    atomicAdd(s + 1, v.y);
    atomicAdd(s + 2, v.z);
    atomicAdd(s + 3, v.w);
    if (c == 0) atomicAdd(&cnts[g], 1.0f);
}

__global__ void pool_div_kernel(float* __restrict__ sums, const float* __restrict__ cnts, int total) {
    int tid = blockIdx.x * blockDim.x + threadIdx.x;
    if (tid >= total) return;
    int g = tid >> 8;
    sums[tid] = sums[tid] / fmaxf(cnts[g], 1.0f);
}

// ---------------- fc2: [G,128] @ [128,12] + b ----------------
__global__ void fc2_kernel(const float* __restrict__ g1, const float* __restrict__ W,
                           const float* __restrict__ b, float* __restrict__ out, int G) {
    int tid = blockIdx.x * blockDim.x + threadIdx.x;
    if (tid >= G * 12) return;
    int g = tid / 12;
    int o = tid - g * 12;
    const float* row = g1 + (size_t)g * 128;
    float s = b[o];
#pragma unroll 8
    for (int k = 0; k < 128; ++k) s += row[k] * W[k * 12 + o];
    out[tid] = s;
}

static inline unsigned nblocks(long long t) { return (unsigned)((t + 255) / 256); }

extern "C" void kernel_launch(void* const* d_in, const int* in_sizes, int n_in,
                              void* d_out, int out_size, void* d_ws, size_t ws_size,
                              hipStream_t stream) {
    const float* x    = (const float*)d_in[0];
    const int*   ei   = (const int*)d_in[1];
    const int*   batch= (const int*)d_in[2];
    const float* W1   = (const float*)d_in[3];
    const float* b1   = (const float*)d_in[4];
    const float* W2   = (const float*)d_in[5];
    const float* b2   = (const float*)d_in[6];
    const float* W3   = (const float*)d_in[7];
    const float* b3   = (const float*)d_in[8];
    const float* fc1W = (const float*)d_in[9];
    const float* fc1b = (const float*)d_in[10];
    const float* fc2W = (const float*)d_in[11];
    const float* fc2b = (const float*)d_in[12];
    float* out = (float*)d_out;

    float* ws   = (float*)d_ws;
    float* dinv = ws;                               // N
    float* P    = dinv + N_NODES;                   // N*256 (ping)
    float* Q    = P + (size_t)N_NODES * 256;        // N*256 (pong)
    float* sums = Q + (size_t)N_NODES * 256;        // G*256
    float* cnts = sums + (size_t)G_GRAPHS * 256;    // G
    float* g1   = cnts + G_GRAPHS;                  // G*128

    const int* src = ei;
    const int* dst = ei + E_EDGES;
    const int TB = 256;

    // degree -> dinv
    hipMemsetAsync(dinv, 0, N_NODES * sizeof(float), stream);
    deg_kernel<<<nblocks(E_EDGES), TB, 0, stream>>>(dst, dinv, E_EDGES);
    dinv_kernel<<<nblocks(N_NODES), TB, 0, stream>>>(dinv, N_NODES);

    // layer 1: P[:, :64] = x * W1 ; agg -> Q ; relu
    layer1_kernel<<<nblocks((long long)N_NODES * 64), TB, 0, stream>>>(x, W1, P, N_NODES);
    hipMemsetAsync(Q, 0, (size_t)N_NODES * 64 * sizeof(float), stream);
    edge_agg_kernel<<<nblocks((long long)E_EDGES * 4), TB, 0, stream>>>(src, dst, P, dinv, Q, E_EDGES, 64);
    selfloop_bias_act<<<nblocks((long long)N_NODES * 64), TB, 0, stream>>>(P, dinv, b1, Q, N_NODES, 64, 1);

    // layer 2: P = Q[N,64] @ W2[64,128] ; agg -> Q ; relu
    {
        long long tiles = (long long)(N_NODES >> 5) * (128 >> 5);
        wmma_gemm2x2<<<nblocks(tiles * 32), TB, 0, stream>>>(Q, W2, nullptr, P, N_NODES, 64, 128, 0);
    }
    hipMemsetAsync(Q, 0, (size_t)N_NODES * 128 * sizeof(float), stream);
    edge_agg_kernel<<<nblocks((long long)E_EDGES * 8), TB, 0, stream>>>(src, dst, P, dinv, Q, E_EDGES, 128);
    selfloop_bias_act<<<nblocks((long long)N_NODES * 128), TB, 0, stream>>>(P, dinv, b2, Q, N_NODES, 128, 1);

    // layer 3: P = Q[N,128] @ W3[128,256] ; agg -> Q (no relu)
    {
        long long tiles = (long long)(N_NODES >> 5) * (256 >> 5);
        wmma_gemm2x2<<<nblocks(tiles * 32), TB, 0, stream>>>(Q, W3, nullptr, P, N_NODES, 128, 256, 0);
    }
    hipMemsetAsync(Q, 0, (size_t)N_NODES * 256 * sizeof(float), stream);
    edge_agg_kernel<<<nblocks((long long)E_EDGES * 16), TB, 0, stream>>>(src, dst, P, dinv, Q, E_EDGES, 256);
    selfloop_bias_act<<<nblocks((long long)N_NODES * 256), TB, 0, stream>>>(P, dinv, b3, Q, N_NODES, 256, 0);

    // global mean pool: sums[G,256], cnts[G]
    hipMemsetAsync(sums, 0, ((size_t)G_GRAPHS * 256 + G_GRAPHS) * sizeof(float), stream);
    pool_sum_kernel<<<nblocks((long long)N_NODES * 64), TB, 0, stream>>>(Q, batch, sums, cnts, N_NODES);
    pool_div_kernel<<<nblocks((long long)G_GRAPHS * 256), TB, 0, stream>>>(sums, cnts, G_GRAPHS * 256);

    // fc1: g1 = relu(sums[G,256] @ fc1W[256,128] + fc1b)
    {
        long long tiles = (long long)(G_GRAPHS >> 5) * (128 >> 5);
        wmma_gemm2x2<<<nblocks(tiles * 32), TB, 0, stream>>>(sums, fc1W, fc1b, g1, G_GRAPHS, 256, 128, 1);
    }

    // fc2: out = g1 @ fc2W[128,12] + fc2b
    fc2_kernel<<<nblocks((long long)G_GRAPHS * 12), TB, 0, stream>>>(g1, fc2W, fc2b, out, G_GRAPHS);
}